// CosineSimCodebook_88888643158513
// MI455X (gfx1250) — compile-verified
//
#include <hip/hip_runtime.h>
#include <hip/hip_bf16.h>
#include <math.h>

// ---------------------------------------------------------------------------
// CosineSimCodebook on MI455X (gfx1250, wave32)
//   x:     [8,4096,512] -> X [M=32768, K=512] f32
//   embed: [1,4096,512] -> E [C=4096,  K=512] f32
// outputs (flat, concatenated f32):
//   quantize [16777216] | embed_ind [32768] | dist [134217728]
//   | new_embed [2097152] | new_cluster_size [4096]
// ---------------------------------------------------------------------------

typedef float v2f __attribute__((ext_vector_type(2)));
typedef float v8f __attribute__((ext_vector_type(8)));

constexpr int M_ROWS = 32768;
constexpr int C_CB   = 4096;
constexpr int D_K    = 512;

constexpr int BM = 128, BN = 128, BK = 16;
constexpr int LDT = 20;  // LDS tile stride (floats): 16B-alignable, 16 distinct banks,
                         // k vs k+2 halves land on disjoint bank classes (mod 4)

constexpr size_t Q_OFF    = 0;
constexpr size_t IND_OFF  = 16777216;
constexpr size_t DIST_OFF = 16809984;
constexpr size_t NE_OFF   = 151027712;
constexpr size_t CS_OFF   = 153124864;

constexpr float DECAY = 0.8f;
constexpr float EPSV  = 1e-5f;

__device__ __forceinline__ unsigned int f32_ordered(float f) {
  const unsigned int u = __float_as_uint(f);
  return (u & 0x80000000u) ? ~u : (u | 0x80000000u);
}

// ---------------------------------------------------------------------------
// K1: dist = X @ E^T via V_WMMA_F32_16X16X4_F32, fused per-row argmax partials.
//   256 threads = 8 wave32; 128x128 tile; BK=16 through LDS; wave grid 4(M)x2(N),
//   each wave 32x64 = 2x4 frags of 16x16. Software-pipelined staging kept
//   strictly in named VGPRs (no lambdas/arrays -> no promote-alloca to LDS).
// ---------------------------------------------------------------------------
__global__ __launch_bounds__(256) void vq_dist_gemm(
    const float* __restrict__ X, const float* __restrict__ E,
    float* __restrict__ dist, unsigned long long* __restrict__ packed) {
  __shared__ float As[BM * LDT];  // [m][k], stride 20
  __shared__ float Bs[BN * LDT];  // [n][k], stride 20 (B = E^T -> store E rows)

  const int tid  = threadIdx.x;
  const int lane = tid & 31;
  const int wave = tid >> 5;
  const int half = lane >> 4;   // 0: lanes 0-15, 1: lanes 16-31
  const int lr   = lane & 15;
  const int mWave = (wave & 3) * 32;
  const int nWave = (wave >> 2) * 64;
  const int rowBase = blockIdx.y * BM;
  const int colBase = blockIdx.x * BN;

  // ---- staging: thread t covers tile rows r0 and r0+64, float4 quad q4
  const int r0 = tid >> 2;
  const int q4 = (tid & 3) * 4;
  const float* xp0 = X + (size_t)(rowBase + r0) * D_K + q4;
  const float* xp1 = xp0 + (size_t)64 * D_K;
  const float* ep0 = E + (size_t)(colBase + r0) * D_K + q4;
  const float* ep1 = ep0 + (size_t)64 * D_K;
  float* as0 = &As[r0 * LDT + q4];
  float* as1 = as0 + 64 * LDT;
  float* bs0 = &Bs[r0 * LDT + q4];
  float* bs1 = bs0 + 64 * LDT;

  // ---- fragment LDS base pointers (contiguous k-pair per lane -> b64 loads)
  const float* aB0 = &As[(mWave + lr) * LDT + half * 2];
  const float* aB1 = aB0 + 16 * LDT;
  const float* bB0 = &Bs[(nWave + lr) * LDT + half * 2];
  const float* bB1 = bB0 + 16 * LDT;
  const float* bB2 = bB0 + 32 * LDT;
  const float* bB3 = bB0 + 48 * LDT;

  v8f acc[2][4];
#pragma unroll
  for (int mb = 0; mb < 2; ++mb)
#pragma unroll
    for (int nb = 0; nb < 4; ++nb)
      acc[mb][nb] = {};

  float4 ra0 = *reinterpret_cast<const float4*>(xp0);
  float4 ra1 = *reinterpret_cast<const float4*>(xp1);
  float4 rb0 = *reinterpret_cast<const float4*>(ep0);
  float4 rb1 = *reinterpret_cast<const float4*>(ep1);

  for (int kt = 0; kt < D_K; kt += BK) {
    *reinterpret_cast<float4*>(as0) = ra0;
    *reinterpret_cast<float4*>(as1) = ra1;
    *reinterpret_cast<float4*>(bs0) = rb0;
    *reinterpret_cast<float4*>(bs1) = rb1;
    __syncthreads();
    if (kt + BK < D_K) {  // next tile's global loads overlap the 32 WMMAs
      ra0 = *reinterpret_cast<const float4*>(xp0 + kt + BK);
      ra1 = *reinterpret_cast<const float4*>(xp1 + kt + BK);
      rb0 = *reinterpret_cast<const float4*>(ep0 + kt + BK);
      rb1 = *reinterpret_cast<const float4*>(ep1 + kt + BK);
    }

#pragma unroll
    for (int ks = 0; ks < BK; ks += 4) {
      // A 16x4: lanes0-15 -> k,k+1 ; lanes16-31 -> k+2,k+3
      const v2f a0 = *reinterpret_cast<const v2f*>(aB0 + ks);
      const v2f a1 = *reinterpret_cast<const v2f*>(aB1 + ks);
      // B 4x16 symmetric layout: per-lane column n, contiguous k pair
      const v2f b0 = *reinterpret_cast<const v2f*>(bB0 + ks);
      const v2f b1 = *reinterpret_cast<const v2f*>(bB1 + ks);
      const v2f b2 = *reinterpret_cast<const v2f*>(bB2 + ks);
      const v2f b3 = *reinterpret_cast<const v2f*>(bB3 + ks);
      acc[0][0] = __builtin_amdgcn_wmma_f32_16x16x4_f32(false, a0, false, b0, (short)0, acc[0][0], false, false);
      acc[0][1] = __builtin_amdgcn_wmma_f32_16x16x4_f32(false, a0, false, b1, (short)0, acc[0][1], false, false);
      acc[0][2] = __builtin_amdgcn_wmma_f32_16x16x4_f32(false, a0, false, b2, (short)0, acc[0][2], false, false);
      acc[0][3] = __builtin_amdgcn_wmma_f32_16x16x4_f32(false, a0, false, b3, (short)0, acc[0][3], false, false);
      acc[1][0] = __builtin_amdgcn_wmma_f32_16x16x4_f32(false, a1, false, b0, (short)0, acc[1][0], false, false);
      acc[1][1] = __builtin_amdgcn_wmma_f32_16x16x4_f32(false, a1, false, b1, (short)0, acc[1][1], false, false);
      acc[1][2] = __builtin_amdgcn_wmma_f32_16x16x4_f32(false, a1, false, b2, (short)0, acc[1][2], false, false);
      acc[1][3] = __builtin_amdgcn_wmma_f32_16x16x4_f32(false, a1, false, b3, (short)0, acc[1][3], false, false);
    }
    __syncthreads();
  }

  // -------- write C frags (lanes 0-15 write consecutive n -> coalesced)
#pragma unroll
  for (int mb = 0; mb < 2; ++mb) {
#pragma unroll
    for (int nb = 0; nb < 4; ++nb) {
      const int n = colBase + nWave + nb * 16 + lr;
#pragma unroll
      for (int v = 0; v < 8; ++v) {
        const int m = rowBase + mWave + mb * 16 + half * 8 + v;
        dist[(size_t)m * C_CB + n] = acc[mb][nb][v];
      }
    }
  }

  // -------- fused per-row argmax over this block's 64 columns per wave-group.
  // Key = (ordered_float << 32) | ~col : atomicMax is order-independent and
  // ties resolve to the smallest column (reference first-index semantics).
#pragma unroll
  for (int mb = 0; mb < 2; ++mb) {
#pragma unroll
    for (int v = 0; v < 8; ++v) {
      const int m = rowBase + mWave + mb * 16 + half * 8 + v;
      float bv = acc[mb][0][v];
      int   bc = colBase + nWave + lr;
#pragma unroll
      for (int nb = 1; nb < 4; ++nb) {
        const float ov = acc[mb][nb][v];
        const int   oc = colBase + nWave + nb * 16 + lr;
        if (ov > bv || (ov == bv && oc < bc)) { bv = ov; bc = oc; }
      }
#pragma unroll
      for (int off = 8; off > 0; off >>= 1) {  // reduce the 16-lane half
        const float ov = __shfl_xor(bv, off, 16);
        const int   oc = __shfl_xor(bc, off, 16);
        if (ov > bv || (ov == bv && oc < bc)) { bv = ov; bc = oc; }
      }
      if (lr == 0) {
        const unsigned long long key =
            ((unsigned long long)f32_ordered(bv) << 32) |
            (unsigned long long)(unsigned int)(~bc);
        atomicMax(&packed[m], key);
      }
    }
  }
}

// ---------------------------------------------------------------------------
// K2: decode packed argmax -> ind (int ws) + embed_ind output (f32)
// ---------------------------------------------------------------------------
__global__ __launch_bounds__(256) void vq_decode(
    const unsigned long long* __restrict__ packed, int* __restrict__ ind,
    float* __restrict__ ind_f) {
  const int i = blockIdx.x * 256 + threadIdx.x;
  if (i < M_ROWS) {
    const int c = (int)(~(unsigned int)(packed[i] & 0xffffffffull));
    ind[i] = c;
    ind_f[i] = (float)c;
  }
}

// ---------------------------------------------------------------------------
// K3: quantize = E[ind]  (float4 gather)
// ---------------------------------------------------------------------------
__global__ __launch_bounds__(256) void vq_gather(
    const float* __restrict__ E, const int* __restrict__ ind,
    float* __restrict__ q) {
  const int gid = blockIdx.x * 256 + threadIdx.x;  // one float4 each
  const int row = gid >> 7;                        // 128 float4 per row
  const int qd  = gid & 127;
  const float4 v = *reinterpret_cast<const float4*>(
      &E[(size_t)ind[row] * D_K + qd * 4]);
  *reinterpret_cast<float4*>(&q[(size_t)row * D_K + qd * 4]) = v;
}

// ---------------------------------------------------------------------------
// K0: zero scratch (packed | bins | embed_sum | tot)
// ---------------------------------------------------------------------------
__global__ __launch_bounds__(256) void vq_zero(float* __restrict__ p, int n) {
  const int i = blockIdx.x * 256 + threadIdx.x;
  if (i < n) p[i] = 0.0f;
}

// ---------------------------------------------------------------------------
// K4: bins += onehot ; embed_sum[idx] += x_row   (one block per row)
// ---------------------------------------------------------------------------
__global__ __launch_bounds__(256) void vq_scatter(
    const float* __restrict__ X, const int* __restrict__ ind,
    float* __restrict__ bins, float* __restrict__ esum) {
  const int row = blockIdx.x;
  const int idx = ind[row];
  if (threadIdx.x == 0) atomicAdd(&bins[idx], 1.0f);
  const float* xr = X + (size_t)row * D_K;
  float* er = esum + (size_t)idx * D_K;
  for (int k = threadIdx.x; k < D_K; k += 256) atomicAdd(&er[k], xr[k]);
}

// ---------------------------------------------------------------------------
// K5: new_cluster_size = cs*DECAY + bins*(1-DECAY); tot = sum(ncs)
// ---------------------------------------------------------------------------
__global__ __launch_bounds__(256) void vq_ema_cs(
    const float* __restrict__ cs, const float* __restrict__ bins,
    float* __restrict__ ncs_ws, float* __restrict__ ncs_out,
    float* __restrict__ tot) {
  __shared__ float red[256];
  const int j = blockIdx.x * 256 + threadIdx.x;  // 16 blocks cover c=4096
  const float v = cs[j] * DECAY + bins[j] * (1.0f - DECAY);
  ncs_ws[j] = v;
  ncs_out[j] = v;
  red[threadIdx.x] = v;
  __syncthreads();
  for (int s = 128; s > 0; s >>= 1) {
    if (threadIdx.x < s) red[threadIdx.x] += red[threadIdx.x + s];
    __syncthreads();
  }
  if (threadIdx.x == 0) atomicAdd(tot, red[0]);
}

// ---------------------------------------------------------------------------
// K6: new_embed = l2norm( (embed_avg*DECAY + embed_sum*(1-DECAY)) / cs_smoothed )
// ---------------------------------------------------------------------------
__global__ __launch_bounds__(256) void vq_ema_embed(
    const float* __restrict__ eavg, const float* __restrict__ esum,
    const float* __restrict__ ncs, const float* __restrict__ tot,
    float* __restrict__ ne) {
  __shared__ float red[256];
  __shared__ float sinv;
  const int j = blockIdx.x;
  const float t = *tot;
  const float cssm = (ncs[j] + EPSV) / (t + (float)C_CB * EPSV) * t;
  const float inv_cs = 1.0f / cssm;
  float v0, v1;
  {
    const int k0 = threadIdx.x;
    const int k1 = threadIdx.x + 256;
    v0 = (eavg[(size_t)j * D_K + k0] * DECAY + esum[(size_t)j * D_K + k0] * (1.0f - DECAY)) * inv_cs;
    v1 = (eavg[(size_t)j * D_K + k1] * DECAY + esum[(size_t)j * D_K + k1] * (1.0f - DECAY)) * inv_cs;
  }
  red[threadIdx.x] = v0 * v0 + v1 * v1;
  __syncthreads();
  for (int s = 128; s > 0; s >>= 1) {
    if (threadIdx.x < s) red[threadIdx.x] += red[threadIdx.x + s];
    __syncthreads();
  }
  if (threadIdx.x == 0) {
    const float nrm = sqrtf(red[0]);
    sinv = 1.0f / fmaxf(nrm, 1e-12f);
  }
  __syncthreads();
  ne[(size_t)j * D_K + threadIdx.x] = v0 * sinv;
  ne[(size_t)j * D_K + threadIdx.x + 256] = v1 * sinv;
}

// ---------------------------------------------------------------------------
extern "C" void kernel_launch(void* const* d_in, const int* in_sizes, int n_in,
                              void* d_out, int out_size, void* d_ws, size_t ws_size,
                              hipStream_t stream) {
  const float* x     = (const float*)d_in[0];  // [32768, 512]
  const float* embed = (const float*)d_in[1];  // [4096, 512]
  const float* cs    = (const float*)d_in[2];  // [4096]
  const float* eavg  = (const float*)d_in[3];  // [4096, 512]

  float* out = (float*)d_out;
  float* q_out    = out + Q_OFF;
  float* ind_outf = out + IND_OFF;
  float* dist_out = out + DIST_OFF;
  float* ne_out   = out + NE_OFF;
  float* cs_out   = out + CS_OFF;

  // ws layout: [packed u64 32768][bins f 4096][esum f 2097152][tot f 1][ncs f 4096][ind i32 32768]
  unsigned long long* packed = (unsigned long long*)d_ws;
  float* wf   = (float*)(packed + M_ROWS);
  float* bins = wf;
  float* esum = wf + C_CB;
  float* tot  = esum + (size_t)C_CB * D_K;
  float* ncs  = tot + 1;
  int*   ind  = (int*)(ncs + C_CB);

  // 0) zero: packed (as 2*M f32) + bins + esum + tot
  {
    const int nz = 2 * M_ROWS + C_CB + C_CB * D_K + 1;
    vq_zero<<<(nz + 255) / 256, 256, 0, stream>>>((float*)d_ws, nz);
  }

  // 1) dist GEMM (WMMA f32 16x16x4) + fused argmax partials
  vq_dist_gemm<<<dim3(C_CB / BN, M_ROWS / BM), 256, 0, stream>>>(x, embed, dist_out, packed);

  // 2) decode argmax
  vq_decode<<<(M_ROWS + 255) / 256, 256, 0, stream>>>(packed, ind, ind_outf);

  // 3) quantize gather
  vq_gather<<<(M_ROWS * (D_K / 4)) / 256, 256, 0, stream>>>(embed, ind, q_out);

  // 4) scatter stats
  vq_scatter<<<M_ROWS, 256, 0, stream>>>(x, ind, bins, esum);

  // 5) EMA cluster size + total
  vq_ema_cs<<<C_CB / 256, 256, 0, stream>>>(cs, bins, ncs, cs_out, tot);

  // 6) EMA embed + laplace smoothing + l2norm
  vq_ema_embed<<<C_CB, 256, 0, stream>>>(eavg, esum, ncs, tot, ne_out);
}